// MemNet_65807488909785
// MI455X (gfx1250) — compile-verified
//
#include <hip/hip_runtime.h>
#include <math.h>

typedef __attribute__((ext_vector_type(2))) float v2f;
typedef __attribute__((ext_vector_type(8))) float v8f;

#define BS    128
#define STORY 100
#define SLEN  20
#define QLEN  30
#define EMB   300
#define NANS  4000

// ---------------------------------------------------------------------------
// Kernel 1: u[b][d] = sum_j emb[q[b][j]][d]        (4.6 MB gather)
// ---------------------------------------------------------------------------
__global__ void memnet_sum_q(const int* __restrict__ q,
                             const float* __restrict__ emb,
                             float* __restrict__ u) {
    int b = blockIdx.x, t = threadIdx.x;
    if (t >= EMB) return;
    float acc = 0.f;
    for (int j = 0; j < QLEN; ++j) {
        int tok = q[b * QLEN + j];
        acc += emb[(size_t)tok * EMB + t];
    }
    u[b * EMB + t] = acc;
}

// ---------------------------------------------------------------------------
// Kernel 2: m[b][s][d] = sum_j emb[x[b][s][j]][d]  (307 MB gather: the roofline)
// Coalesced over d; one block per (b,s) sentence.
// ---------------------------------------------------------------------------
__global__ void memnet_sum_x(const int* __restrict__ x,
                             const float* __restrict__ emb,
                             float* __restrict__ m) {
    int bs = blockIdx.x, t = threadIdx.x;
    if (t >= EMB) return;
    float acc = 0.f;
    for (int j = 0; j < SLEN; ++j) {
        int tok = x[bs * SLEN + j];
        acc += emb[(size_t)tok * EMB + t];
    }
    m[(size_t)bs * EMB + t] = acc;
}

// ---------------------------------------------------------------------------
// Kernel 3 (x3 steps): one block per batch row.
//   p = softmax(m[b] @ u[b]);  u[b] += p @ m[b]
// m is L2-resident (15.4 MB << 192 MB); u and p live in LDS.
// ---------------------------------------------------------------------------
__global__ void memnet_hop(const float* __restrict__ m,
                           float* __restrict__ u) {
    __shared__ float u_s[EMB];
    __shared__ float p_s[STORY];
    int b = blockIdx.x, t = threadIdx.x;

    if (t < EMB) u_s[t] = u[b * EMB + t];
    __syncthreads();

    if (t < STORY) {
        const float* row = m + (size_t)(b * STORY + t) * EMB;
        float acc = 0.f;
        for (int d = 0; d < EMB; ++d) acc += row[d] * u_s[d];
        p_s[t] = acc;
    }
    __syncthreads();

    if (t == 0) {   // softmax over 100 entries: cheap serial pass
        float mx = p_s[0];
        for (int s = 1; s < STORY; ++s) mx = fmaxf(mx, p_s[s]);
        float sum = 0.f;
        for (int s = 0; s < STORY; ++s) { float e = expf(p_s[s] - mx); p_s[s] = e; sum += e; }
        float inv = 1.f / sum;
        for (int s = 0; s < STORY; ++s) p_s[s] *= inv;
    }
    __syncthreads();

    if (t < EMB) {  // o[d] = sum_s p[s]*m[b,s,d]; coalesced over d
        float o = 0.f;
        for (int s = 0; s < STORY; ++s)
            o += p_s[s] * m[(size_t)(b * STORY + s) * EMB + t];
        u[b * EMB + t] = u_s[t] + o;
    }
}

// ---------------------------------------------------------------------------
// Kernel 4: sim[128,4000] = u[128,300] @ glove[4000,300]^T via f32 WMMA.
// One wave per 16x16 tile; 75 chained V_WMMA_F32_16X16X4_F32 (K=4) per tile.
// A 16x4 layout (ISA 7.12.2): lanes 0-15 -> M=lane, K={0,1}; lanes 16-31 -> K={2,3}.
// B 4x16 symmetric with N on lanes. C/D: VGPR r -> row r (lanes 0-15) / r+8.
// ---------------------------------------------------------------------------
__global__ void memnet_sim_wmma(const float* __restrict__ u,
                                const float* __restrict__ glove,
                                float* __restrict__ sim) {
    int nbase = blockIdx.x * 16;   // answer tile
    int mbase = blockIdx.y * 16;   // batch tile
    int lane  = threadIdx.x;
    int half  = lane >> 4;         // 0: K pair {0,1}, 1: K pair {2,3}
    int idx   = lane & 15;         // M (for A) / N (for B)

    const float* arow = u     + (size_t)(mbase + idx) * EMB;
    const float* brow = glove + (size_t)(nbase + idx) * EMB;

    v8f c = {};
    for (int k0 = 0; k0 < EMB; k0 += 4) {           // 300 = 75 * 4, exact
        int kb = k0 + half * 2;
        v2f a;  a.x  = arow[kb]; a.y  = arow[kb + 1];
        v2f bb; bb.x = brow[kb]; bb.y = brow[kb + 1];
        c = __builtin_amdgcn_wmma_f32_16x16x4_f32(
                /*neg_a=*/false, a, /*neg_b=*/false, bb,
                /*c_mod=*/(short)0, c, /*reuse_a=*/false, /*reuse_b=*/false);
    }

    float* dst = sim + (size_t)(mbase + half * 8) * NANS + nbase + idx;
#pragma unroll
    for (int r = 0; r < 8; ++r)
        dst[(size_t)r * NANS] = c[r];
}

// ---------------------------------------------------------------------------
// Kernel 5: log_softmax over each row of sim -> out
// ---------------------------------------------------------------------------
__global__ void memnet_logsoftmax(const float* __restrict__ sim,
                                  float* __restrict__ out) {
    __shared__ float red[256];
    int b = blockIdx.x, t = threadIdx.x;
    const float* row = sim + (size_t)b * NANS;

    float mx = -INFINITY;
    for (int i = t; i < NANS; i += 256) mx = fmaxf(mx, row[i]);
    red[t] = mx; __syncthreads();
    for (int s = 128; s > 0; s >>= 1) {
        if (t < s) red[t] = fmaxf(red[t], red[t + s]);
        __syncthreads();
    }
    mx = red[0]; __syncthreads();

    float sum = 0.f;
    for (int i = t; i < NANS; i += 256) sum += expf(row[i] - mx);
    red[t] = sum; __syncthreads();
    for (int s = 128; s > 0; s >>= 1) {
        if (t < s) red[t] += red[t + s];
        __syncthreads();
    }
    float lse = logf(red[0]) + mx;

    for (int i = t; i < NANS; i += 256)
        out[(size_t)b * NANS + i] = row[i] - lse;
}

// ---------------------------------------------------------------------------
extern "C" void kernel_launch(void* const* d_in, const int* in_sizes, int n_in,
                              void* d_out, int out_size, void* d_ws, size_t ws_size,
                              hipStream_t stream) {
    const int*   q     = (const int*)d_in[0];     // [128,1,30]
    const int*   x     = (const int*)d_in[1];     // [128,100,20]
    const float* emb   = (const float*)d_in[2];   // [400000,300]
    const float* glove = (const float*)d_in[3];   // [4000,300]
    float*       out   = (float*)d_out;           // [128,4000]

    // workspace partition (~17.6 MB total)
    float* u   = (float*)d_ws;                    // BS*EMB
    float* m   = u + (size_t)BS * EMB;            // BS*STORY*EMB
    float* sim = m + (size_t)BS * STORY * EMB;    // BS*NANS

    memnet_sum_q<<<BS, 320, 0, stream>>>(q, emb, u);
    memnet_sum_x<<<BS * STORY, 320, 0, stream>>>(x, emb, m);
    for (int step = 0; step < 3; ++step)
        memnet_hop<<<BS, 320, 0, stream>>>(m, u);

    dim3 gsim(NANS / 16, BS / 16);                // 250 x 8 tiles, 1 wave each
    memnet_sim_wmma<<<gsim, 32, 0, stream>>>(u, glove, sim);

    memnet_logsoftmax<<<BS, 256, 0, stream>>>(sim, out);
}